// DiverseSiblingsSearch_1202590843068
// MI455X (gfx1250) — compile-verified
//
#include <hip/hip_runtime.h>
#include <hip/hip_bf16.h>
#include <stdint.h>

// Problem constants (fixed by the harness)
#define BSZ   64
#define BEAM  8
#define VOCAB 50257
#define STEPDIM 5
#define KTOP  16
#define DIVERSITY_RATE 0.5f

// Stage-1 tiling
#define S1_THREADS 256
#define S1_WAVES   8            // 256 / wave32
#define CHUNK      256          // floats per chunk per wave
#define ISSUES     (CHUNK / 32) // async b32 issues per chunk (8)
#define NCHUNKS    ((VOCAB + CHUNK - 1) / CHUNK) // 197

// ---- CDNA5 async global->LDS path (probe-guarded) -------------------------
#if defined(__gfx1250__) && __has_builtin(__builtin_amdgcn_global_load_async_to_lds_b32) && __has_builtin(__builtin_amdgcn_s_wait_asynccnt)
#define HAVE_ASYNC_LDS 1
#else
#define HAVE_ASYNC_LDS 0
#endif

__device__ __forceinline__ void async_ld_b32(void* ldsDst, const void* gSrc) {
#if HAVE_ASYNC_LDS
  typedef __attribute__((address_space(3))) int lds_i32_t;
  typedef __attribute__((address_space(1))) int glb_i32_t;
  // flat->LDS: low-32-bit truncation yields the LDS offset (LLVM's own lowering)
  lds_i32_t* l = (lds_i32_t*)(unsigned int)(uintptr_t)ldsDst;
  glb_i32_t* g = (glb_i32_t*)(uintptr_t)gSrc;
  __builtin_amdgcn_global_load_async_to_lds_b32(g, l, 0, 0);
#else
  *(float*)ldsDst = __builtin_nontemporal_load((const float*)gSrc);
#endif
}

#if HAVE_ASYNC_LDS
#define WAIT_ASYNC(n) do { __builtin_amdgcn_s_wait_asynccnt(n); asm volatile("" ::: "memory"); } while (0)
#else
#define WAIT_ASYNC(n) do { asm volatile("" ::: "memory"); } while (0)
#endif

// ---- per-thread sorted top-K insert (registers, fully unrolled) -----------
__device__ __forceinline__ void topk_insert(float (&v)[KTOP], int (&ix)[KTOP],
                                            float nv, int ni) {
  if (nv <= v[KTOP - 1]) return;   // early-out: almost always taken
  v[KTOP - 1] = nv; ix[KTOP - 1] = ni;
  #pragma unroll
  for (int j = KTOP - 1; j > 0; --j) {
    if (v[j] > v[j - 1]) {
      float tv = v[j]; v[j] = v[j - 1]; v[j - 1] = tv;
      int   ti = ix[j]; ix[j] = ix[j - 1]; ix[j - 1] = ti;
    }
  }
}

// ---------------------------------------------------------------------------
// Stage 1: one block per (bsz,beam) row. Async-stream the row through LDS,
// per-thread register top-16, then log2(256)-level sorted-list merge tree.
// Writes penalized sorted scores + vocab indices to workspace.
// ---------------------------------------------------------------------------
__global__ __launch_bounds__(S1_THREADS) void dss_stage1(
    const float* __restrict__ lprobs, const float* __restrict__ scores,
    const int* __restrict__ stepPtr, float* __restrict__ ws_s,
    int* __restrict__ ws_i) {
  __shared__ float sbuf[S1_WAVES][2][CHUNK];      // 16 KB streaming buffers
  __shared__ float lv[S1_THREADS * KTOP];         // 16 KB merge values
  __shared__ int   li[S1_THREADS * KTOP];         // 16 KB merge indices

  const int row  = blockIdx.x;                    // 0..511
  const int tid  = threadIdx.x;
  const int wave = tid >> 5;
  const int lane = tid & 31;
  const float* __restrict__ src = lprobs + (size_t)row * VOCAB;

  float tv[KTOP]; int ti[KTOP];
  #pragma unroll
  for (int q = 0; q < KTOP; ++q) { tv[q] = -__builtin_inff(); ti[q] = 0; }

  // ---- streaming: wave w handles chunks w, w+8, ... with double buffer ----
  int c = wave;                                   // always < NCHUNKS (197)
  #pragma unroll
  for (int j = 0; j < ISSUES; ++j) {
    int gi = c * CHUNK + j * 32 + lane;
    int gc = gi < VOCAB ? gi : (VOCAB - 1);       // clamp (ignored in scan)
    async_ld_b32(&sbuf[wave][0][j * 32 + lane], src + gc);
  }
  int buf = 0;
  while (c < NCHUNKS) {
    const int cn = c + S1_WAVES;
    if (cn < NCHUNKS) {
      #pragma unroll
      for (int j = 0; j < ISSUES; ++j) {
        int gi = cn * CHUNK + j * 32 + lane;
        int gc = gi < VOCAB ? gi : (VOCAB - 1);
        async_ld_b32(&sbuf[wave][buf ^ 1][j * 32 + lane], src + gc);
      }
      WAIT_ASYNC(ISSUES);   // older chunk complete (async loads finish in order)
    } else {
      WAIT_ASYNC(0);
    }
    #pragma unroll
    for (int j = 0; j < ISSUES; ++j) {
      int gi = c * CHUNK + j * 32 + lane;
      float v = sbuf[wave][buf][j * 32 + lane];
      if (gi < VOCAB) topk_insert(tv, ti, v, gi);
    }
    buf ^= 1;
    c = cn;
  }

  // ---- publish per-thread sorted lists, then pairwise top-16 merge tree ---
  #pragma unroll
  for (int q = 0; q < KTOP; ++q) { lv[tid * KTOP + q] = tv[q]; li[tid * KTOP + q] = ti[q]; }
  __syncthreads();

  for (int stride = 1; stride < S1_THREADS; stride <<= 1) {
    if ((tid & (2 * stride - 1)) == 0) {
      const int a = tid * KTOP, b = (tid + stride) * KTOP;
      float mv[KTOP]; int mi[KTOP];
      int i = 0, j = 0;
      #pragma unroll
      for (int q = 0; q < KTOP; ++q) {
        float va = lv[a + i], vb = lv[b + j];
        bool ta = (va >= vb);
        mv[q] = ta ? va : vb;
        mi[q] = ta ? li[a + i] : li[b + j];
        if (ta) ++i; else ++j;
      }
      #pragma unroll
      for (int q = 0; q < KTOP; ++q) { lv[a + q] = mv[q]; li[a + q] = mi[q]; }
    }
    __syncthreads();
  }

  // ---- bias + sibling penalty applied only to the winning 16 --------------
  if (tid < KTOP) {
    const int st = stepPtr[0];
    float bias = 0.f, pen = 0.f;
    if (st != 0) {
      bias = scores[row * STEPDIM + (st - 1)];
      pen  = (float)(tid + 1) * DIVERSITY_RATE;
    }
    ws_s[row * KTOP + tid] = lv[tid] + bias - pen;   // still sorted descending
    ws_i[row * KTOP + tid] = li[tid];
  }
}

// ---------------------------------------------------------------------------
// Stage 2: one block per bsz. Merge 8 sorted per-beam 16-lists -> top-16 of
// 128, tracking beam + vocab index. Writes (scores | indices | beams).
// ---------------------------------------------------------------------------
__global__ __launch_bounds__(64) void dss_stage2(
    const float* __restrict__ ws_s, const int* __restrict__ ws_i,
    const int* __restrict__ stepPtr, float* __restrict__ out) {
  __shared__ float lv[BEAM * KTOP];
  __shared__ int   lb[BEAM * KTOP];
  __shared__ int   lx[BEAM * KTOP];

  const int b = blockIdx.x, tid = threadIdx.x;
  const int st = stepPtr[0];

  if (tid < BEAM) {
    for (int q = 0; q < KTOP; ++q) {
      lv[tid * KTOP + q] = ws_s[(b * BEAM + tid) * KTOP + q];
      lx[tid * KTOP + q] = ws_i[(b * BEAM + tid) * KTOP + q];
      lb[tid * KTOP + q] = tid;
    }
  }
  __syncthreads();

  if (st != 0) {
    for (int stride = 1; stride < BEAM; stride <<= 1) {
      if (tid < BEAM && (tid & (2 * stride - 1)) == 0) {
        const int a = tid * KTOP, bb = (tid + stride) * KTOP;
        float mv[KTOP]; int mbm[KTOP], mix[KTOP];
        int i = 0, j = 0;
        #pragma unroll
        for (int q = 0; q < KTOP; ++q) {
          float va = lv[a + i], vb = lv[bb + j];
          bool ta = (va >= vb);
          mv[q]  = ta ? va : vb;
          mbm[q] = ta ? lb[a + i] : lb[bb + j];
          mix[q] = ta ? lx[a + i] : lx[bb + j];
          if (ta) ++i; else ++j;
        }
        #pragma unroll
        for (int q = 0; q < KTOP; ++q) { lv[a+q]=mv[q]; lb[a+q]=mbm[q]; lx[a+q]=mix[q]; }
      }
      __syncthreads();
    }
  }
  // st==0: beam-0 list already sits in slots [0,16) with beams==0, no penalty.

  if (tid < KTOP) {
    out[b * KTOP + tid]                   = lv[tid];                 // final_scores
    out[BSZ * KTOP + b * KTOP + tid]      = (float)lx[tid];          // final_indices
    out[2 * BSZ * KTOP + b * KTOP + tid]  = (float)(st == 0 ? 0 : lb[tid]); // final_beams
  }
}

extern "C" void kernel_launch(void* const* d_in, const int* in_sizes, int n_in,
                              void* d_out, int out_size, void* d_ws, size_t ws_size,
                              hipStream_t stream) {
  (void)in_sizes; (void)n_in; (void)out_size; (void)ws_size;
  const float* lprobs = (const float*)d_in[0];   // (64, 8, 50257) f32
  const float* scores = (const float*)d_in[1];   // (64, 8, 5)     f32
  const int*   step   = (const int*)d_in[2];     // scalar (=5)
  float* out  = (float*)d_out;                   // 3 x (64,16) concatenated
  float* ws_s = (float*)d_ws;
  int*   ws_i = (int*)((char*)d_ws + (size_t)BSZ * BEAM * KTOP * sizeof(float));

  dss_stage1<<<BSZ * BEAM, S1_THREADS, 0, stream>>>(lprobs, scores, step, ws_s, ws_i);
  dss_stage2<<<BSZ, 64, 0, stream>>>(ws_s, ws_i, step, out);
}